// NumFeatureExtractor_28595892257097
// MI455X (gfx1250) — compile-verified
//
#include <hip/hip_runtime.h>
#include <stdint.h>

// Problem constants from the reference setup.
#define F_DIM   64
#define H_DIM   768
#define HV4     192            // H_DIM / 4
#define LN_EPS  1e-5f
#define EMB_EPS 1e-12f

// Native 4-float vector (clang ext vector) — required by
// __builtin_nontemporal_store (HIP's float4 is a struct and is rejected).
typedef __attribute__((ext_vector_type(4))) float v4f;

// ---------------------------------------------------------------------------
// CDNA5 async global->LDS copy (gfx1250). Each active lane copies 16 bytes
// from its global address to its LDS byte offset. Tracked by ASYNCcnt.
// Generic pointers to __shared__ carry the LDS byte offset in their low 32
// bits (flat aperture mapping).
// ---------------------------------------------------------------------------
__device__ __forceinline__ void async_ld_b128(uint32_t lds_byte_off, const void* gptr) {
  asm volatile("global_load_async_to_lds_b128 %0, %1, off"
               :: "v"(lds_byte_off), "v"((unsigned long long)(uintptr_t)gptr)
               : "memory");
}
__device__ __forceinline__ void wait_async_zero() {
  asm volatile("s_wait_asynccnt 0" ::: "memory");
}

// ---------------------------------------------------------------------------
// Kernel 1: per-feature precompute (batch independent).
//   c[f,h]  = b[f,h] + pos[f,h] + tok0[h]
//   Wg[f,h] = W[f,h] * g[h],  Cg[f,h] = c[f,h] * g[h]
//   moments[f] = { mW, mC, A2=Var(W), A1=Cov(W,c), A0=Var(c) }  (over h)
// 64 blocks (one per f) x 192 threads; thread t owns float4 chunk h=4t.
// ---------------------------------------------------------------------------
__global__ __launch_bounds__(192)
void nfe_precompute(const float* __restrict__ W, const float* __restrict__ bias,
                    const float* __restrict__ pos, const float* __restrict__ tok,
                    const float* __restrict__ g,
                    float* __restrict__ Wg, float* __restrict__ Cg,
                    float* __restrict__ mom) {
  const int f = blockIdx.x;
  const int t = threadIdx.x;

  const v4f w4 = ((const v4f*)(W    + f * H_DIM))[t];
  const v4f b4 = ((const v4f*)(bias + f * H_DIM))[t];
  const v4f p4 = ((const v4f*)(pos  + f * H_DIM))[t];
  const v4f k4 = ((const v4f*)tok)[t];                  // tok_emb row 0
  const v4f g4 = ((const v4f*)g)[t];

  const v4f c4 = b4 + p4 + k4;
  ((v4f*)Wg)[f * HV4 + t] = w4 * g4;
  ((v4f*)Cg)[f * HV4 + t] = c4 * g4;

  float sw  = w4.x + w4.y + w4.z + w4.w;
  float sc  = c4.x + c4.y + c4.z + c4.w;
  float sww = w4.x*w4.x + w4.y*w4.y + w4.z*w4.z + w4.w*w4.w;
  float swc = w4.x*c4.x + w4.y*c4.y + w4.z*c4.z + w4.w*c4.w;
  float scc = c4.x*c4.x + c4.y*c4.y + c4.z*c4.z + c4.w*c4.w;

  // wave32 shuffle reduction, then cross-wave via LDS (6 waves)
#pragma unroll
  for (int off = 16; off > 0; off >>= 1) {
    sw  += __shfl_down(sw,  off);
    sc  += __shfl_down(sc,  off);
    sww += __shfl_down(sww, off);
    swc += __shfl_down(swc, off);
    scc += __shfl_down(scc, off);
  }
  __shared__ float red[6][5];
  const int wave = t >> 5, lane = t & 31;
  if (lane == 0) {
    red[wave][0] = sw;  red[wave][1] = sc;  red[wave][2] = sww;
    red[wave][3] = swc; red[wave][4] = scc;
  }
  __syncthreads();
  if (t == 0) {
    float SW = 0.f, SC = 0.f, SWW = 0.f, SWC = 0.f, SCC = 0.f;
#pragma unroll
    for (int i = 0; i < 6; ++i) {
      SW += red[i][0]; SC += red[i][1]; SWW += red[i][2];
      SWC += red[i][3]; SCC += red[i][4];
    }
    const float inv = 1.0f / (float)H_DIM;
    const float mW = SW * inv, mC = SC * inv;
    float* m = mom + f * 8;                      // padded to 8 floats
    m[0] = mW;
    m[1] = mC;
    m[2] = SWW * inv - mW * mW;                  // A2 = Var(W)
    m[3] = SWC * inv - mW * mC;                  // A1 = Cov(W,c)
    m[4] = SCC * inv - mC * mC;                  // A0 = Var(c)
  }
}

// ---------------------------------------------------------------------------
// Kernel 2: one block per batch row b. 384 threads (12 wave32s).
//   - async-stage x[b,0:64] into LDS (CDNA5 async-to-LDS path)
//   - LayerNorm over F with wave32 shuffles
//   - per-f scalars:  mu2 = xn*mW+mC,  var2 = A2*xn^2 + 2*A1*xn + A0
//   - stream out[b,f,h] = a*Wg + r*Cg - m*g + beta  as NT b128 stores
// Thread t owns h-chunk j = t%192 (loop-invariant g/beta in registers) and
// f-parity t/192; 32 iterations cover f = 0..63.
// Output (402 MB) > L2 (192 MB) and is never re-read -> NT store hint keeps
// the L2-resident Wg/Cg (384 KB, re-read by all 2048 blocks) from eviction.
// ---------------------------------------------------------------------------
__global__ __launch_bounds__(384)
void nfe_fused(const float* __restrict__ x,
               const float* __restrict__ ln_g, const float* __restrict__ ln_b,
               const float* __restrict__ Wg, const float* __restrict__ Cg,
               const float* __restrict__ mom,
               const float* __restrict__ g, const float* __restrict__ beta,
               float* __restrict__ out) {
  const int b = blockIdx.x;
  const int t = threadIdx.x;

  __shared__ float xrow[F_DIM];
  __shared__ float sA[F_DIM], sR[F_DIM], sM[F_DIM];
  __shared__ float stats[2];

  // Stage the 256-byte x row via async DMA: 16 lanes x b128.
  if (t < 16) {
    const uint32_t lds_off = (uint32_t)(uintptr_t)(&xrow[t * 4]);
    async_ld_b128(lds_off, x + (size_t)b * F_DIM + t * 4);
  }
  wait_async_zero();
  __syncthreads();

  // LayerNorm stats over F=64 within wave 0.
  if (t < 32) {
    const float v0 = xrow[t], v1 = xrow[t + 32];
    float s = v0 + v1;
    float q = v0 * v0 + v1 * v1;
#pragma unroll
    for (int off = 16; off > 0; off >>= 1) {
      s += __shfl_down(s, off);
      q += __shfl_down(q, off);
    }
    if (t == 0) {
      const float mu  = s * (1.0f / F_DIM);
      const float var = q * (1.0f / F_DIM) - mu * mu;
      stats[0] = mu;
      stats[1] = rsqrtf(var + LN_EPS);
    }
  }
  __syncthreads();

  // Per-f scalars for the (analytically folded) embedding LayerNorm.
  if (t < F_DIM) {
    const float mu = stats[0], rs = stats[1];
    const float xn = fmaf((xrow[t] - mu) * rs, ln_g[t], ln_b[t]);
    const float* m = mom + t * 8;
    const float mu2  = fmaf(xn, m[0], m[1]);
    const float var2 = fmaf(fmaf(m[2], xn, 2.0f * m[3]), xn, m[4]);
    const float r2   = rsqrtf(var2 + EMB_EPS);
    sA[t] = xn * r2;
    sR[t] = r2;
    sM[t] = mu2 * r2;
  }
  __syncthreads();

  const int j  = t % HV4;          // float4 chunk within the H dimension
  const int fh = t / HV4;          // 0 or 1 -> f parity
  const v4f g4  = ((const v4f*)g)[j];
  const v4f be4 = ((const v4f*)beta)[j];
  const v4f* __restrict__ Wg4 = (const v4f*)Wg;
  const v4f* __restrict__ Cg4 = (const v4f*)Cg;
  v4f* __restrict__ out4 = (v4f*)out + (size_t)b * (F_DIM * HV4);

#pragma unroll 4
  for (int k = 0; k < 32; ++k) {
    const int f = 2 * k + fh;
    const float a = sA[f], r = sR[f], m = sM[f];
    const int idx = f * HV4 + j;
    const v4f w = Wg4[idx];
    const v4f c = Cg4[idx];
    if (k < 31) {                                   // hint next L2-resident rows
      __builtin_prefetch(&Wg4[idx + 2 * HV4], 0, 1);
      __builtin_prefetch(&Cg4[idx + 2 * HV4], 0, 1);
    }
    v4f o;
    o.x = fmaf(a, w.x, fmaf(r, c.x, fmaf(-m, g4.x, be4.x)));
    o.y = fmaf(a, w.y, fmaf(r, c.y, fmaf(-m, g4.y, be4.y)));
    o.z = fmaf(a, w.z, fmaf(r, c.z, fmaf(-m, g4.z, be4.z)));
    o.w = fmaf(a, w.w, fmaf(r, c.w, fmaf(-m, g4.w, be4.w)));
    __builtin_nontemporal_store(o, &out4[idx]);     // global_store_b128 th:NT
  }
}

// ---------------------------------------------------------------------------
extern "C" void kernel_launch(void* const* d_in, const int* in_sizes, int n_in,
                              void* d_out, int out_size, void* d_ws, size_t ws_size,
                              hipStream_t stream) {
  const float* x     = (const float*)d_in[0];   // [B,64]
  const float* ln_g  = (const float*)d_in[1];   // [64]
  const float* ln_b  = (const float*)d_in[2];   // [64]
  const float* W     = (const float*)d_in[3];   // [64,768]
  const float* bias  = (const float*)d_in[4];   // [64,768]
  const float* pos   = (const float*)d_in[5];   // [512,768] (rows 0..63 used)
  const float* tok   = (const float*)d_in[6];   // [2,768]   (row 0 used)
  const float* eg    = (const float*)d_in[7];   // [768]
  const float* eb    = (const float*)d_in[8];   // [768]
  float* out = (float*)d_out;

  // Workspace layout (~395 KB): Wg | Cg | moments(64x8)
  float* Wg  = (float*)d_ws;
  float* Cg  = Wg + F_DIM * H_DIM;
  float* mom = Cg + F_DIM * H_DIM;

  const int B = in_sizes[0] / F_DIM;            // 2048

  nfe_precompute<<<F_DIM, 192, 0, stream>>>(W, bias, pos, tok, eg, Wg, Cg, mom);
  nfe_fused<<<B, 384, 0, stream>>>(x, ln_g, ln_b, Wg, Cg, mom, eg, eb, out);
}